// RadialFunction_13778255086019
// MI455X (gfx1250) — compile-verified
//
#include <hip/hip_runtime.h>

// ---------------------------------------------------------------------------
// RadialFunction: per-pair Gaussian radial basis * gathered species embedding
//   out[n, r] = cutoff(dr[n]) * EMBED_NORM * sum_b E[Zj,Zi,r,b] * basis(dr[n])_b
//
// Roofline: 12 B/pair stream-in + 20 B/pair stream-out = 256 MB HBM at
// 23.3 TB/s (~11 us floor). The 2 MB embedding table lives in the 192 MB L2;
// NT cache hints keep the 256 MB of single-touch streams from evicting it.
// WMMA deliberately unused: per-pair unique 5x7 matrices give <2% tile
// utilization, and the kernel is store-bound, not compute-bound.
//
// v2: wave-local LDS transpose (no __syncthreads; per-wave DS ordering is
// guaranteed in-order), exp in log2 domain (native v_exp_f32), cutoff via
// revolution-domain v_cos_f32 (no range reduction).
// ---------------------------------------------------------------------------

#define N_RADIAL 5
#define N_BASIS 7
#define N_SPECIES 119

__constant__ float c_shifts[N_BASIS] = {
    0.5f, 1.28571428571f, 2.07142857143f, 2.85714285714f,
    3.64285714286f, 4.42857142857f, 5.21428571429f};

constexpr float R_MAX       = 6.0f;
// BETTA * log2(e) : basis_b = exp2(-B2 * d^2)
constexpr float B2          = 1.9636682222764441f;   // (49/36)*1.4426950408889634
constexpr float SCALE_CONST = 0.36466562f;           // RAD_NORM * EMBED_NORM
                                                     // 0.9648176 * 0.37796447
constexpr float INV_2RMAX   = 1.0f / 12.0f;          // cos(pi*d/6)=cos(2pi*d/12)

constexpr int WAVE  = 32;
constexpr int BLOCK = 256;   // 8 wave32s
constexpr int ITEMS = 4;     // pairs per thread (prefetch pipeline depth)
constexpr int TILE  = BLOCK * ITEMS;

__device__ __forceinline__ void compute_pair(float dr,
                                             const float* __restrict__ row,
                                             float* __restrict__ res) {
  // Gaussian basis in log2 domain: v_exp_f32 TRANS ops co-execute with VALU
  float basis[N_BASIS];
#pragma unroll
  for (int b = 0; b < N_BASIS; ++b) {
    float d  = c_shifts[b] - dr;
    basis[b] = __builtin_amdgcn_exp2f(-B2 * (d * d));
  }
  // cosine cutoff: hardware cos in revolutions, arg in [0, 0.5]
  float drc    = fminf(dr, R_MAX);
  float cutoff = 0.5f * (__builtin_amdgcn_cosf(drc * INV_2RMAX) + 1.0f);
  float scale  = SCALE_CONST * cutoff;   // folds RAD_NORM * EMBED_NORM
  // 5x7 contraction (35 FMAs); rows serviced from L2-resident table
#pragma unroll
  for (int r = 0; r < N_RADIAL; ++r) {
    float acc = 0.0f;
#pragma unroll
    for (int b = 0; b < N_BASIS; ++b)
      acc = fmaf(row[r * N_BASIS + b], basis[b], acc);
    res[r] = acc * scale;
  }
}

__global__ __launch_bounds__(BLOCK) void radial_fn_kernel(
    const float* __restrict__ dr_g, const int* __restrict__ Zi_g,
    const int* __restrict__ Zj_g, const float* __restrict__ emb,
    float* __restrict__ out, int n) {
  // Per-wave private transpose slices: no cross-wave traffic, no barriers.
  __shared__ float lds[BLOCK * N_RADIAL];

  const int  tid      = threadIdx.x;
  const int  lane     = tid & (WAVE - 1);
  const int  waveBase = (tid & ~(WAVE - 1)) * N_RADIAL;  // waveId * 160
  const long tile     = (long)blockIdx.x * TILE;

  if (tile + TILE <= (long)n) {
    // ---- stage 1: coalesced NT stream loads + gather-row prefetch ----
    float        d[ITEMS];
    const float* row[ITEMS];
#pragma unroll
    for (int t = 0; t < ITEMS; ++t) {
      long i  = tile + (long)t * BLOCK + tid;
      d[t]    = __builtin_nontemporal_load(&dr_g[i]);   // single-touch: th:NT
      int zi  = __builtin_nontemporal_load(&Zi_g[i]);
      int zj  = __builtin_nontemporal_load(&Zj_g[i]);
      row[t]  = emb + (size_t)(zj * N_SPECIES + zi) * (N_RADIAL * N_BASIS);
      __builtin_prefetch(row[t], 0, 3);                 // global_prefetch_b8
    }
    // ---- stage 2: compute, wave-local LDS transpose, coalesced NT stores --
#pragma unroll
    for (int t = 0; t < ITEMS; ++t) {
      float res[N_RADIAL];
      compute_pair(d[t], row[t], res);
#pragma unroll
      for (int r = 0; r < N_RADIAL; ++r)
        lds[waveBase + lane * N_RADIAL + r] = res[r];   // stride 5: no conflict
      // Same-wave LDS ops are in-order; no barrier needed.
      long outBase = (tile + (long)t * BLOCK) * N_RADIAL + waveBase;
#pragma unroll
      for (int k = 0; k < N_RADIAL; ++k) {
        int j = k * WAVE + lane;                        // unit stride per wave
        __builtin_nontemporal_store(lds[waveBase + j], &out[outBase + j]);
      }
    }
  } else {
    // ---- tail tile: guarded scalar path, direct NT stores ----
    for (int t = 0; t < ITEMS; ++t) {
      long i = tile + (long)t * BLOCK + tid;
      if (i < (long)n) {
        int zi = Zi_g[i];
        int zj = Zj_g[i];
        const float* row =
            emb + (size_t)(zj * N_SPECIES + zi) * (N_RADIAL * N_BASIS);
        float res[N_RADIAL];
        compute_pair(dr_g[i], row, res);
#pragma unroll
        for (int r = 0; r < N_RADIAL; ++r)
          __builtin_nontemporal_store(res[r], &out[i * N_RADIAL + r]);
      }
    }
  }
}

extern "C" void kernel_launch(void* const* d_in, const int* in_sizes, int n_in,
                              void* d_out, int out_size, void* d_ws,
                              size_t ws_size, hipStream_t stream) {
  // setup_inputs order: dr [N], Z_i [N], Z_j [N], embeddings [119,119,5,7]
  const float* dr  = (const float*)d_in[0];
  const int*   Zi  = (const int*)d_in[1];
  const int*   Zj  = (const int*)d_in[2];
  const float* emb = (const float*)d_in[3];
  float*       out = (float*)d_out;

  const int n      = in_sizes[0];
  const int blocks = (n + TILE - 1) / TILE;
  radial_fn_kernel<<<blocks, BLOCK, 0, stream>>>(dr, Zi, Zj, emb, out, n);
}